// MAMDense_86148454023833
// MI455X (gfx1250) — compile-verified
//
#include <hip/hip_runtime.h>
#include <stdint.h>

// MAM dense: C[m,n] = max_k(x[m,k]*w[n,k]) + min_k(x[m,k]*w[n,k]) + bias[n]
// x: [4096,1024] f32, w: [1024,1024] f32 (N,K), bias: [1024] f32, C: [4096,1024] f32

#define M_DIM 4096
#define K_DIM 1024
#define N_DIM 1024

#define BM 128
#define BN 64
#define BK 32
#define TM 8
#define TN 4
#define BMP (BM + 4) /* 132 dwords: k-major A row stride, 16B aligned, depads banks */
#define BNP (BN + 4) /* 68 dwords */
#define NTHREADS 256
#define KTILES (K_DIM / BK)

// Low 32 bits of a generic pointer to LDS == byte offset from the wave's LDS base
// (flat->LDS mapping truncates to addr[31:0]).
__device__ __forceinline__ uint32_t lds_off32(const void* p) {
  return (uint32_t)(uintptr_t)p;
}

// Async gather: per-lane global dword -> per-lane LDS dword (tracked by ASYNCcnt).
// VDST carries the LDS byte address, ADDR the 64-bit global address.
__device__ __forceinline__ void async_gather_b32(uint32_t lds_byte_off, const float* gaddr) {
  asm volatile("global_load_async_to_lds_b32 %0, %1, off"
               :: "v"(lds_byte_off), "v"(gaddr)
               : "memory");
}

__device__ __forceinline__ void wait_async0() {
  asm volatile("s_wait_asynccnt 0" ::: "memory");
}

__global__ __launch_bounds__(NTHREADS, 2)
void mam_dense_kernel(const float* __restrict__ x,
                      const float* __restrict__ w,
                      const float* __restrict__ bias,
                      float* __restrict__ c) {
  __shared__ __align__(16) float As[2][BK * BMP];
  __shared__ __align__(16) float Bs[2][BK * BNP];

  const int t     = threadIdx.x;
  const int mBase = blockIdx.y * BM;
  const int nBase = blockIdx.x * BN;

  // Copy mapping: lane group covers k (coalesced 128B global reads),
  // scatters into k-major LDS (transpose for free).
  const int kk = t & 31;  // 0..31 : k within tile
  const int rr = t >> 5;  // 0..7  : row group

  // ---- prologue: stage tile 0 ----
  {
    const float* xg = x + (size_t)mBase * K_DIM + kk;
    const float* wg = w + (size_t)nBase * K_DIM + kk;
#pragma unroll
    for (int i = 0; i < 16; ++i) {
      const int m = rr + i * 8;  // 0..127
      async_gather_b32(lds_off32(&As[0][kk * BMP + m]), xg + (size_t)m * K_DIM);
    }
#pragma unroll
    for (int i = 0; i < 8; ++i) {
      const int n = rr + i * 8;  // 0..63
      async_gather_b32(lds_off32(&Bs[0][kk * BNP + n]), wg + (size_t)n * K_DIM);
    }
  }
  wait_async0();
  __syncthreads();

  // Compute mapping: 16x16 thread grid over the 128x64 tile.
  const int tx = t & 15;
  const int ty = t >> 4;
  const int m0 = ty * TM;
  const int n0 = tx * TN;

  float vmax[TM][TN], vmin[TM][TN];
#pragma unroll
  for (int i = 0; i < TM; ++i)
#pragma unroll
    for (int j = 0; j < TN; ++j) {
      vmax[i][j] = -__builtin_inff();
      vmin[i][j] =  __builtin_inff();
    }

  for (int kt = 0; kt < KTILES; ++kt) {
    const int cur = kt & 1;

    // Issue async copy of the NEXT tile; it overlaps this tile's compute.
    if (kt + 1 < KTILES) {
      const int nxt = cur ^ 1;
      const int k0  = (kt + 1) * BK;
      const float* xg = x + (size_t)mBase * K_DIM + k0 + kk;
      const float* wg = w + (size_t)nBase * K_DIM + k0 + kk;
#pragma unroll
      for (int i = 0; i < 16; ++i) {
        const int m = rr + i * 8;
        async_gather_b32(lds_off32(&As[nxt][kk * BMP + m]), xg + (size_t)m * K_DIM);
      }
#pragma unroll
      for (int i = 0; i < 8; ++i) {
        const int n = rr + i * 8;
        async_gather_b32(lds_off32(&Bs[nxt][kk * BNP + n]), wg + (size_t)n * K_DIM);
      }
    }

    const float* Ak = &As[cur][0];
    const float* Bk = &Bs[cur][0];

    // Two k-steps per iteration -> v_max3_num_f32 / v_min3_num_f32 folding.
#pragma unroll
    for (int k2 = 0; k2 < BK; k2 += 2) {
      const float4 a00 = *(const float4*)&Ak[(k2    ) * BMP + m0];
      const float4 a01 = *(const float4*)&Ak[(k2    ) * BMP + m0 + 4];
      const float4 a10 = *(const float4*)&Ak[(k2 + 1) * BMP + m0];
      const float4 a11 = *(const float4*)&Ak[(k2 + 1) * BMP + m0 + 4];
      const float4 b0  = *(const float4*)&Bk[(k2    ) * BNP + n0];
      const float4 b1  = *(const float4*)&Bk[(k2 + 1) * BNP + n0];

      const float a0v[TM] = {a00.x, a00.y, a00.z, a00.w, a01.x, a01.y, a01.z, a01.w};
      const float a1v[TM] = {a10.x, a10.y, a10.z, a10.w, a11.x, a11.y, a11.z, a11.w};
      const float b0v[TN] = {b0.x, b0.y, b0.z, b0.w};
      const float b1v[TN] = {b1.x, b1.y, b1.z, b1.w};

#pragma unroll
      for (int i = 0; i < TM; ++i)
#pragma unroll
        for (int j = 0; j < TN; ++j) {
          const float p0 = a0v[i] * b0v[j];
          const float p1 = a1v[i] * b1v[j];
          vmax[i][j] = fmaxf(fmaxf(vmax[i][j], p0), p1);
          vmin[i][j] = fminf(fminf(vmin[i][j], p0), p1);
        }
    }

    wait_async0();     // next-tile copy done (per wave)
    __syncthreads();   // all waves' copies visible; safe to flip buffers
  }

  // ---- epilogue: C = vmax + vmin + bias ----
  const float4 bv = *(const float4*)&bias[nBase + n0];
  const float bvv[TN] = {bv.x, bv.y, bv.z, bv.w};
#pragma unroll
  for (int i = 0; i < TM; ++i) {
    float4 out;
    out.x = vmax[i][0] + vmin[i][0] + bvv[0];
    out.y = vmax[i][1] + vmin[i][1] + bvv[1];
    out.z = vmax[i][2] + vmin[i][2] + bvv[2];
    out.w = vmax[i][3] + vmin[i][3] + bvv[3];
    *(float4*)&c[(size_t)(mBase + m0 + i) * N_DIM + nBase + n0] = out;
  }
}

extern "C" void kernel_launch(void* const* d_in, const int* in_sizes, int n_in,
                              void* d_out, int out_size, void* d_ws, size_t ws_size,
                              hipStream_t stream) {
  const float* x    = (const float*)d_in[0];
  const float* wt   = (const float*)d_in[1];
  const float* bias = (const float*)d_in[2];
  float* c          = (float*)d_out;

  dim3 grid(N_DIM / BN, M_DIM / BM);  // (16, 32) = 512 workgroups
  mam_dense_kernel<<<grid, NTHREADS, 0, stream>>>(x, wt, bias, c);
}